// Module_54631984005192
// MI455X (gfx1250) — compile-verified
//
#include <hip/hip_runtime.h>
#include <cstdint>

#define BATCH 16
#define SEQ   8192
#define DIM   512
#define NH    8
#define NHPAD 16          /* qkeff padded to 16 head rows (8..15 zero) */
#define HD    64
#define LSPLIT 8
#define LCHUNK (SEQ / LSPLIT)
#define LN_EPS 1e-5f
#define SCALE 0.25f       /* 1/(sqrt(HD)*TAU) = 1/(8*0.5) */

typedef __attribute__((ext_vector_type(2))) float v2f;
typedef __attribute__((ext_vector_type(8))) float v8f;

// ---------------------------------------------------------------------------
// 0) zero-fill helper (zrow: dummy zero weight row for padded WMMA lanes)
// ---------------------------------------------------------------------------
__global__ void zero_fill_kernel(float* __restrict__ p, int n) {
    const int i = blockIdx.x * blockDim.x + threadIdx.x;
    if (i < n) p[i] = 0.f;
}

// ---------------------------------------------------------------------------
// 1) q = Q @ Wq + bq      (16 x 512, tiny)
// ---------------------------------------------------------------------------
__global__ void prep_q_kernel(const float* __restrict__ Q, const float* __restrict__ Wq,
                              const float* __restrict__ bq, float* __restrict__ q) {
    const int b = blockIdx.x, d = threadIdx.x;
    const float* Qr = Q + b * DIM;
    float acc = bq[d];
#pragma unroll 8
    for (int c = 0; c < DIM; ++c) acc += Qr[c] * Wq[c * DIM + d];
    q[b * DIM + d] = acc;
}

// ---------------------------------------------------------------------------
// 2) qk_eff[b,hh,c] = SCALE * sum_j Wk[c, h*64+j] * q[b, h*64+j]   (hh<8)
//                   = 0                                            (hh>=8, pad)
//    sbias[b,h]     = SCALE * sum_j bk[h*64+j]    * q[b, h*64+j]
// ---------------------------------------------------------------------------
__global__ void prep_qkeff_kernel(const float* __restrict__ q, const float* __restrict__ Wk,
                                  const float* __restrict__ bk,
                                  float* __restrict__ qkeff, float* __restrict__ sbias) {
    const int bh = blockIdx.x;            // 0 .. BATCH*NHPAD-1
    const int b = bh >> 4, hh = bh & 15;
    const int c = threadIdx.x;
    if (hh >= NH) {                       // zero padding rows -> unconditional WMMA loads
        qkeff[(size_t)bh * DIM + c] = 0.f;
        return;
    }
    const float* qh = q + b * DIM + hh * HD;
    float acc = 0.f;
#pragma unroll 8
    for (int j = 0; j < HD; ++j) acc += Wk[c * DIM + hh * HD + j] * qh[j];
    qkeff[(size_t)bh * DIM + c] = acc * SCALE;
    if (c == 0) {
        float sb = 0.f;
        for (int j = 0; j < HD; ++j) sb += bk[hh * HD + j] * qh[j];
        sbias[b * NH + hh] = sb * SCALE;
    }
}

// ---------------------------------------------------------------------------
// 3) scores[b,h,l] = K[b,l,:].qk_eff[b,h,:] + sbias  — streams K (256 MB, NT)
//    WMMA f32 16x16x4: M = 16 L-rows, N = 16 (8 heads + zero pad), K-dim = 512
//    grid (SEQ/128, BATCH) x 256 thr; wave w owns rows lbase = bx*128 + w*16
// ---------------------------------------------------------------------------
__global__ void scores_kernel(const float* __restrict__ K, const float* __restrict__ qkeff,
                              const float* __restrict__ sbias, float* __restrict__ scores) {
    const int b    = blockIdx.y;
    const int wave = threadIdx.x >> 5;
    const int lane = threadIdx.x & 31;
    const int m    = lane & 15;               // A: M row ; B: N col
    const int koff = (lane >> 4) << 1;        // 0 or 2
    const int lbase = blockIdx.x * 128 + wave * 16;

    const float* Krow = K + ((size_t)b * SEQ + (lbase + m)) * DIM + koff;
    const float* Bq   = qkeff + (size_t)(b * NHPAD + m) * DIM + koff;  // rows 8..15 == 0

    v8f acc = {};
#pragma unroll 8
    for (int k = 0; k < DIM; k += 4) {
        v2f a  = __builtin_nontemporal_load((const v2f*)(Krow + k));   // streamed once
        v2f bb = *(const v2f*)(Bq + k);                                // hot, cached
        acc = __builtin_amdgcn_wmma_f32_16x16x4_f32(false, a, false, bb, (short)0, acc,
                                                    false, false);
    }
    if (m < NH) {  // D: VGPR r -> row (r + (lane>=16)*8), col = lane&15 = head
        const float sb = sbias[b * NH + m];
        float* srow = scores + (size_t)(b * NH + m) * SEQ + lbase + ((lane >> 4) << 3);
#pragma unroll
        for (int r = 0; r < 8; ++r) srow[r] = acc[r] + sb;
    }
}

// ---------------------------------------------------------------------------
// 4) softmax over L per (b,h), in place
// ---------------------------------------------------------------------------
__global__ void softmax_kernel(float* __restrict__ scores) {
    float* s = scores + (size_t)blockIdx.x * SEQ;
    __shared__ float red[256];
    const int t = threadIdx.x;
    float mx = -3.4e38f;
    for (int i = t; i < SEQ; i += 256) mx = fmaxf(mx, s[i]);
    red[t] = mx; __syncthreads();
    for (int off = 128; off > 0; off >>= 1) {
        if (t < off) red[t] = fmaxf(red[t], red[t + off]);
        __syncthreads();
    }
    mx = red[0]; __syncthreads();
    float sum = 0.f;
    for (int i = t; i < SEQ; i += 256) sum += __expf(s[i] - mx);
    red[t] = sum; __syncthreads();
    for (int off = 128; off > 0; off >>= 1) {
        if (t < off) red[t] += red[t + off];
        __syncthreads();
    }
    const float inv = 1.0f / red[0];
    for (int i = t; i < SEQ; i += 256) s[i] = __expf(s[i] - mx) * inv;
}

// ---------------------------------------------------------------------------
// 5) Upart[z][b,h,c] = sum_{l in chunk z} w[b,h,l] * V[b,l,c] — streams V (NT)
//    WMMA f32 16x16x4: M = 8 heads (pad 16 via zrow), N = 16 cols, K = L chunk
//    grid (DIM/128, BATCH, LSPLIT) x 256 thr
// ---------------------------------------------------------------------------
__global__ void wv_kernel(const float* __restrict__ V, const float* __restrict__ wgt,
                          const float* __restrict__ zrow, float* __restrict__ Upart) {
    const int b    = blockIdx.y;
    const int z    = blockIdx.z;
    const int wave = threadIdx.x >> 5;
    const int lane = threadIdx.x & 31;
    const int n    = lane & 15;               // A: head row ; B: col within tile
    const int koff = (lane >> 4) << 1;
    const int c0   = blockIdx.x * 128 + wave * 16;
    const int l0   = z * LCHUNK;

    // uniform per-lane pointer select OUTSIDE the loop: no divergence inside
    const float* wrow = (n < NH) ? (wgt + (size_t)(b * NH + n) * SEQ + l0 + koff)
                                 : (zrow + l0 + koff);
    const float* vcol = V + ((size_t)b * SEQ + l0 + koff) * DIM + c0 + n;

    v8f acc = {};
#pragma unroll 8
    for (int l = 0; l < LCHUNK; l += 4) {
        v2f a = *(const v2f*)(wrow + l);                               // hot, cached
        v2f bb;
        bb.x = __builtin_nontemporal_load(vcol + (size_t)l * DIM);     // streamed once
        bb.y = __builtin_nontemporal_load(vcol + (size_t)(l + 1) * DIM);
        acc = __builtin_amdgcn_wmma_f32_16x16x4_f32(false, a, false, bb, (short)0, acc,
                                                    false, false);
    }
    if (lane < 16) {  // rows M=0..7 (valid heads) live in lanes 0-15, VGPR r = head
#pragma unroll
        for (int r = 0; r < 8; ++r)
            Upart[((size_t)z * (BATCH * NH) + (b * NH + r)) * DIM + c0 + n] = acc[r];
    }
}

// ---------------------------------------------------------------------------
// 6) U = sum_z Upart[z]   (fixed order -> deterministic)
// ---------------------------------------------------------------------------
__global__ void reduce_u_kernel(const float* __restrict__ Upart, float* __restrict__ U) {
    const int i = blockIdx.x * blockDim.x + threadIdx.x;   // < 16*8*512
    float acc = 0.f;
#pragma unroll
    for (int z = 0; z < LSPLIT; ++z) acc += Upart[(size_t)z * (BATCH * NH * DIM) + i];
    U[i] = acc;
}

// ---------------------------------------------------------------------------
// 7) ctx = U @ Wv(+bv) per head slice; out = ctx @ Wo + bo; LayerNorm; +Q
// ---------------------------------------------------------------------------
__global__ void finalize_kernel(const float* __restrict__ U, const float* __restrict__ Wv,
                                const float* __restrict__ bv, const float* __restrict__ Wo,
                                const float* __restrict__ bo, const float* __restrict__ gamma,
                                const float* __restrict__ beta, const float* __restrict__ Q,
                                float* __restrict__ out) {
    const int b = blockIdx.x, t = threadIdx.x;
    __shared__ float ctx[DIM];
    __shared__ float red[DIM];
    const int h = t >> 6;
    const float* u = U + (size_t)(b * NH + h) * DIM;
    float acc = bv[t];
#pragma unroll 8
    for (int c = 0; c < DIM; ++c) acc += u[c] * Wv[c * DIM + t];
    ctx[t] = acc;
    __syncthreads();
    float o = bo[t];
#pragma unroll 8
    for (int c = 0; c < DIM; ++c) o += ctx[c] * Wo[c * DIM + t];
    // LayerNorm over DIM
    red[t] = o; __syncthreads();
    for (int off = 256; off > 0; off >>= 1) {
        if (t < off) red[t] += red[t + off];
        __syncthreads();
    }
    const float mu = red[0] * (1.0f / DIM);
    __syncthreads();
    const float dv = o - mu;
    red[t] = dv * dv; __syncthreads();
    for (int off = 256; off > 0; off >>= 1) {
        if (t < off) red[t] += red[t + off];
        __syncthreads();
    }
    const float var = red[0] * (1.0f / DIM);
    out[b * DIM + t] = dv * rsqrtf(var + LN_EPS) * gamma[t] + beta[t] + Q[b * DIM + t];
}

// ---------------------------------------------------------------------------
extern "C" void kernel_launch(void* const* d_in, const int* in_sizes, int n_in,
                              void* d_out, int out_size, void* d_ws, size_t ws_size,
                              hipStream_t stream) {
    const float* Q  = (const float*)d_in[0];
    const float* K  = (const float*)d_in[1];
    const float* V  = (const float*)d_in[2];
    const float* Wq = (const float*)d_in[3];
    const float* bq = (const float*)d_in[4];
    const float* Wk = (const float*)d_in[5];
    const float* bk = (const float*)d_in[6];
    const float* Wv = (const float*)d_in[7];
    const float* bv = (const float*)d_in[8];
    const float* Wo = (const float*)d_in[9];
    const float* bo = (const float*)d_in[10];
    const float* gamma = (const float*)d_in[11];
    const float* beta  = (const float*)d_in[12];
    (void)in_sizes; (void)n_in; (void)out_size; (void)ws_size;

    float* ws     = (float*)d_ws;
    float* q      = ws;                                        // 16*512
    float* qkeff  = q + BATCH * DIM;                           // 16*16*512 (padded)
    float* sbias  = qkeff + BATCH * NHPAD * DIM;               // 128 (pad 256)
    float* scores = sbias + 256;                               // 16*8*8192
    float* Upart  = scores + (size_t)BATCH * NH * SEQ;         // 8*128*512
    float* U      = Upart + (size_t)LSPLIT * BATCH * NH * DIM; // 128*512
    float* zrow   = U + BATCH * NH * DIM;                      // 8192 zeros
    float* out    = (float*)d_out;

    zero_fill_kernel<<<SEQ / 256, 256, 0, stream>>>(zrow, SEQ);
    prep_q_kernel<<<BATCH, DIM, 0, stream>>>(Q, Wq, bq, q);
    prep_qkeff_kernel<<<BATCH * NHPAD, DIM, 0, stream>>>(q, Wk, bk, qkeff, sbias);
    scores_kernel<<<dim3(SEQ / 128, BATCH), 256, 0, stream>>>(K, qkeff, sbias, scores);
    softmax_kernel<<<BATCH * NH, 256, 0, stream>>>(scores);
    wv_kernel<<<dim3(DIM / 128, BATCH, LSPLIT), 256, 0, stream>>>(V, scores, zrow, Upart);
    reduce_u_kernel<<<(BATCH * NH * DIM) / 512, 512, 0, stream>>>(Upart, U);
    finalize_kernel<<<BATCH, DIM, 0, stream>>>(U, Wv, bv, Wo, bo, gamma, beta, Q, out);
}